// ComponentwiseFlow_39195871543473
// MI455X (gfx1250) — compile-verified
//
#include <hip/hip_runtime.h>

// ---------------------------------------------------------------------------
// Rational-quadratic spline flow forward + logdet for MI455X (gfx1250).
// N=262144 rows, D=128 dims, K=32 bins -> 33 knots per dim.
// Memory-bound (~269 MB @ 23.3 TB/s ~= 11.5us floor). Knot tables (50.7 KB)
// are staged per-block into LDS (TDM DMA + verified cooperative fill); x/y
// are streamed as float4 with one wave32 per row and an in-wave shuffle
// reduction for the per-row logdet sum.
// ---------------------------------------------------------------------------

constexpr int   Nn   = 262144;
constexpr int   Dd   = 128;
constexpr int   Kk   = 32;
constexpr int   NK   = Kk + 1;          // 33 knots
constexpr int   KTOT = Dd * NK;         // 4224 floats per table
constexpr int   KTOT3 = 3 * KTOT;       // 12672 floats total (x_pos|y_pos|slopes)
constexpr int   THREADS = 256;          // 8 waves -> 8 rows per iteration
constexpr int   ROWS_PER_BLOCK = 512;   // 512 blocks total
constexpr float RMIN = -5.0f;
constexpr float TOTAL = 10.0f;
constexpr float MINBIN = 1e-4f;
constexpr float MINSLOPE = 1e-4f;

typedef unsigned int u32x4 __attribute__((ext_vector_type(4)));
typedef int          i32x4 __attribute__((ext_vector_type(4)));
typedef int          i32x8 __attribute__((ext_vector_type(8)));

// ---------------------------------------------------------------------------
// TDM staging: one 1D DMA of the whole knot table (global -> LDS).
// D# group0: count=1, lds_addr[63:32], global_addr[120:64], type=2[127:126].
// D# group1: data_size=4B, tensor_dim0=tile_dim0=12672, dim1=1, stride0=12672.
// Groups 2/3: dims 2..4 = 1 (in bounds), tiles unused. 5th vector arg
// (clang-23 6-arg form) zero-filled.
// ---------------------------------------------------------------------------
__device__ __forceinline__ void tdm_issue_knots(const float* gk, float* kn) {
#if __has_builtin(__builtin_amdgcn_tensor_load_to_lds)
  unsigned ldsoff = (unsigned)(unsigned long long)kn;   // low 32 bits = LDS byte offset
  unsigned long long ga = (unsigned long long)gk;
  u32x4 g0 = { 1u,
               ldsoff,
               (unsigned)(ga & 0xFFFFFFFFull),
               (unsigned)((ga >> 32) & 0x01FFFFFFull) | 0x80000000u /*type=2*/ };
  i32x8 g1 = { (int)(2u << 16),                          // data_size = 4B, no flags
               (int)(((unsigned)KTOT3 & 0xFFFFu) << 16), // tensor_dim0[15:0]
               (int)((((unsigned)KTOT3 >> 16) & 0xFFFFu) | (1u << 16)), // dim0 hi | tensor_dim1=1
               (int)(((unsigned)KTOT3 & 0xFFFFu) << 16), // tile_dim0
               1,                                        // tile_dim1 = 1
               (int)KTOT3,                               // tensor_dim0_stride
               0, 0 };
  i32x4 g2 = { 1, 1, 0, 0 };                             // tensor_dim2=1, tensor_dim3=1
  i32x4 g3 = { 0, (int)(1u << 16), 0, 0 };               // tensor_dim4=1
  i32x8 g4 = { 0, 0, 0, 0, 0, 0, 0, 0 };                 // extra group (clang-23 form)
  __builtin_amdgcn_tensor_load_to_lds(g0, g1, g2, g3, g4, 0);
#else
  (void)gk; (void)kn;
#endif
}

__device__ __forceinline__ void tdm_wait_all() {
#if __has_builtin(__builtin_amdgcn_s_wait_tensorcnt)
  __builtin_amdgcn_s_wait_tensorcnt(0);
#endif
}

// ---------------------------------------------------------------------------
// Kernel 1: spline params -> knot tables in d_ws.
// gk layout: [0,KTOT)=x_pos[d][33], [KTOT,2K)=y_pos, [2K,3K)=slopes.
// ---------------------------------------------------------------------------
__global__ void rqs_prep(const float* __restrict__ sp, float* __restrict__ gk) {
  int d = threadIdx.x;
  if (d >= Dd) return;
  const float* p = sp + d * (3 * Kk + 1);
  const float span = TOTAL - Kk * MINBIN;

  // widths -> x_pos, heights -> y_pos (softmax with max-subtract, then cumsum)
  for (int which = 0; which < 2; ++which) {
    const float* u = p + which * Kk;
    float m = -INFINITY;
    for (int j = 0; j < Kk; ++j) m = fmaxf(m, u[j]);
    float s = 0.0f;
    for (int j = 0; j < Kk; ++j) s += expf(u[j] - m);
    float inv = 1.0f / s;
    float* out = gk + which * KTOT + d * NK;
    float acc = 0.0f;
    out[0] = RMIN;
    for (int j = 0; j < Kk; ++j) {
      float w = expf(u[j] - m) * inv * span + MINBIN;
      acc += w;
      out[j + 1] = RMIN + acc;
    }
  }

  // slopes = softplus(us + offset) + MINSLOPE, offset = log(expm1(1 - MINSLOPE))
  const float off = logf(expm1f(1.0f - MINSLOPE));
  const float* us = p + 2 * Kk;
  float* so = gk + 2 * KTOT + d * NK;
  for (int j = 0; j < NK; ++j) {
    float t = us[j] + off;
    float spv = (t > 15.0f) ? t : log1pf(expf(t));
    so[j] = spv + MINSLOPE;
  }
}

// ---------------------------------------------------------------------------
// Per-element spline evaluation (knots in LDS).
// ---------------------------------------------------------------------------
__device__ __forceinline__ float rqs_one(float v, const float* __restrict__ xk,
                                         const float* __restrict__ yk,
                                         const float* __restrict__ sk,
                                         float& ld) {
  const float xlo0 = xk[0];
  const float xhiN = xk[Kk];
  if (v <= xlo0) {                       // linear tail below
    float s = sk[0];
    ld = logf(s);
    return (v - xlo0) * s + yk[0];
  }
  if (v >= xhiN) {                       // linear tail above
    float s = sk[Kk];
    ld = logf(s);
    return (v - xhiN) * s + yk[Kk];
  }
  // binary search: largest idx in [0,31] with xk[idx] <= v
  int idx = 0;
  float x0 = xlo0;
#pragma unroll
  for (int st = 16; st >= 1; st >>= 1) {
    int t = idx + st;
    float xt = xk[t];
    if (xt <= v) { idx = t; x0 = xt; }
  }
  float x1 = xk[idx + 1];
  float y0 = yk[idx], y1 = yk[idx + 1];
  float s0 = sk[idx], s1 = sk[idx + 1];
  float bw = x1 - x0;
  float bh = y1 - y0;
  float bs = bh / bw;
  float z = (v - x0) / bw;
  z = fminf(fmaxf(z, 0.0f), 1.0f);
  float sqz  = z * z;
  float z1mz = z - sqz;
  float omz  = 1.0f - z;
  float sq1mz = omz * omz;
  float sterm = s1 + s0 - 2.0f * bs;
  float num = bh * (bs * sqz + s0 * z1mz);
  float den = bs + sterm * z1mz;
  ld = 2.0f * logf(bs) + logf(s1 * sqz + 2.0f * bs * z1mz + s0 * sq1mz)
       - 2.0f * logf(den);
  return y0 + num / den;
}

// ---------------------------------------------------------------------------
// Kernel 2: main streaming kernel. One wave32 per row; lane l handles dims
// 4l..4l+3 as a float4. Logdet reduced with __shfl_xor inside the wave.
// ---------------------------------------------------------------------------
__global__ __launch_bounds__(THREADS) void rqs_main(const float* __restrict__ x,
                                                    const float* __restrict__ gk,
                                                    float* __restrict__ yout,
                                                    float* __restrict__ ldout) {
  __shared__ float kn[KTOT3];

  // Stage knot tables. TDM DMA first (completes before anyone touches LDS),
  // then a cooperative fill of the same region with known-good values so the
  // final LDS contents are correct regardless of D# bit-packing semantics.
  if (threadIdx.x < 32) {
    tdm_issue_knots(gk, kn);
    tdm_wait_all();
  }
  __syncthreads();
  for (int i = threadIdx.x; i < KTOT3; i += THREADS) kn[i] = gk[i];
  __syncthreads();

  const int lane = threadIdx.x & 31;
  const int wv   = threadIdx.x >> 5;             // 0..7 (wave = row within group)
  const float4* __restrict__ x4 = (const float4*)x;
  float4* __restrict__ y4 = (float4*)yout;

  const long long row0 = (long long)blockIdx.x * ROWS_PER_BLOCK + wv;
  constexpr int WAVES = THREADS / 32;            // 8 rows per iteration
  constexpr int ITERS = ROWS_PER_BLOCK / WAVES;  // 64

  for (int r = 0; r < ITERS; ++r) {
    const long long row = row0 + (long long)r * WAVES;
    const size_t base = (size_t)row * (Dd / 4) + lane;
    float4 xv = x4[base];
    if (r + 1 < ITERS)
      __builtin_prefetch((const void*)(x4 + base + (size_t)WAVES * (Dd / 4)), 0, 0);

    float vs[4] = { xv.x, xv.y, xv.z, xv.w };
    float ys[4];
    float ldsum = 0.0f;
#pragma unroll
    for (int j = 0; j < 4; ++j) {
      const int d = lane * 4 + j;
      const float* xk = kn + d * NK;
      const float* yk = xk + KTOT;
      const float* sk = xk + 2 * KTOT;
      float ld;
      ys[j] = rqs_one(vs[j], xk, yk, sk, ld);
      ldsum += ld;
    }
    float4 yv; yv.x = ys[0]; yv.y = ys[1]; yv.z = ys[2]; yv.w = ys[3];
    y4[base] = yv;

#pragma unroll
    for (int m = 16; m >= 1; m >>= 1) ldsum += __shfl_xor(ldsum, m, 32);
    if (lane == 0) ldout[row] = ldsum;
  }
}

// ---------------------------------------------------------------------------
// Harness entry point.
// ---------------------------------------------------------------------------
extern "C" void kernel_launch(void* const* d_in, const int* in_sizes, int n_in,
                              void* d_out, int out_size, void* d_ws, size_t ws_size,
                              hipStream_t stream) {
  (void)in_sizes; (void)n_in; (void)out_size; (void)ws_size;
  const float* x  = (const float*)d_in[0];          // (N, D) f32
  const float* sp = (const float*)d_in[1];          // (D, 97) f32
  float* y  = (float*)d_out;                        // (N, D) f32
  float* ld = y + (size_t)Nn * Dd;                  // (N,)  f32
  float* gk = (float*)d_ws;                         // 3*128*33 floats of knots

  rqs_prep<<<1, 128, 0, stream>>>(sp, gk);
  rqs_main<<<Nn / ROWS_PER_BLOCK, THREADS, 0, stream>>>(x, gk, y, ld);
}